// EvolvingSystem_16432544875257
// MI455X (gfx1250) — compile-verified
//
#include <hip/hip_runtime.h>
#include <hip/hip_bf16.h>

typedef float v2f __attribute__((ext_vector_type(2)));
typedef float v8f __attribute__((ext_vector_type(8)));
typedef unsigned int u32x4 __attribute__((ext_vector_type(4)));
typedef int i32x4 __attribute__((ext_vector_type(4)));
typedef int i32x8 __attribute__((ext_vector_type(8)));

#define B_SZ 8192
#define C_SZ 16
#define D_SZ 256

#if defined(__AMDGCN__) && __has_builtin(__builtin_amdgcn_tensor_load_to_lds) && \
    __has_builtin(__builtin_amdgcn_s_wait_tensorcnt)
#define USE_TDM 1
#warning "CDNA5 probe: TDM path ACTIVE (tensor_load_to_lds + s_wait_tensorcnt)"
#else
#define USE_TDM 0
#if defined(__AMDGCN__)
#warning "CDNA5 probe: TDM builtins NOT available - register-staging fallback"
#endif
#endif

// ---------------------------------------------------------------------------
// Kernel 1: p[c][j] = sum_i mu[c,i] * sigma_inv[c,i,j]   (C*D = 4096 outputs)
// ---------------------------------------------------------------------------
__global__ __launch_bounds__(256) void evo_pvec(const float* __restrict__ mu,
                                                const float* __restrict__ sig,
                                                float* __restrict__ p) {
  const int gt = blockIdx.x * 256 + threadIdx.x;  // 0..4095
  const int c = gt >> 8;
  const int j = gt & 255;
  const float* sc = sig + c * (D_SZ * D_SZ) + j;
  const float* mc = mu + c * D_SZ;
  float acc = 0.0f;
#pragma unroll 8
  for (int i = 0; i < D_SZ; ++i) acc = fmaf(mc[i], sc[i * D_SZ], acc);
  p[gt] = acc;
}

#if USE_TDM
// ---------------------------------------------------------------------------
// Tensor Data Mover: stage one 16x256 fp32 chunk of sigma_inv[c] into LDS,
// re-tiled as [n=16][k=16][col=16] (LDS-sequential write order; global walk:
// X = 16 contiguous floats, Y = 16 rows @ stride 256, Z = 16 blocks @ stride 16).
// D# layout per CDNA5 ISA.
// ---------------------------------------------------------------------------
__device__ __forceinline__ void tdm_load_chunk(const float* gsrc,
                                               unsigned lds_byte_off) {
  const unsigned long long ga = (unsigned long long)(uintptr_t)gsrc;
  u32x4 g0;
  g0[0] = 1u;                                        // count=1 (valid user D#)
  g0[1] = lds_byte_off;                              // lds_addr
  g0[2] = (unsigned)(ga & 0xffffffffull);            // global_addr[31:0]
  g0[3] = (unsigned)((ga >> 32) & 0x01ffffffull) |   // global_addr[56:32]
          (2u << 30);                                // type = 2 ("image")
  i32x8 g1;
  g1[0] = (2 << 16);          // workgroup_mask=0, data_size=4B, no pad/iter
  g1[1] = (int)(256u << 16);  // tensor_dim0 = 256 (low16 @ [31:16])
  g1[2] = (int)(256u << 16);  // tensor_dim0 hi=0 | tensor_dim1 = 256 (low16)
  g1[3] = (int)(16u << 16);   // tensor_dim1 hi=0 | tile_dim0 = 16
  g1[4] = (int)(16u | (16u << 16));  // tile_dim1 = 16 | tile_dim2 = 16
  g1[5] = 256;                // tensor_dim0_stride = 256 (elements)
  g1[6] = (int)(16u << 16);   // stride0 hi=0 | tensor_dim1_stride = 16 (low16)
  g1[7] = 0;                  // tensor_dim1_stride hi = 0
  i32x4 g2 = {16, 0, 0, 0};   // tensor_dim2 = 16; no dim3/iterate
  i32x4 g3 = {0, 0, 0, 0};
#if __clang_major__ >= 23
  i32x8 g4 = {0, 0, 0, 0, 0, 0, 0, 0};
  __builtin_amdgcn_tensor_load_to_lds(g0, g1, g2, g3, g4, 0);
#else
  __builtin_amdgcn_tensor_load_to_lds(g0, g1, g2, g3, 0);
#endif
}
#endif  // USE_TDM

// ---------------------------------------------------------------------------
// Inner compute: 4 K-steps x 16 N-tiles of V_WMMA_F32_16X16X4_F32 on one chunk.
// LDS layout [n][k][16]: B-frag pair = {base + kk*16, base + kk*16 + 16}
// -> single ds_load_2addr into an adjacent VGPR pair, conflict-free banks.
// Software-pipelined: a rotating 16-entry B-frag register file keeps ~16 LDS
// loads in flight for every K-step (issue next-kk4 load before consuming).
// ---------------------------------------------------------------------------
__device__ __forceinline__ void compute_chunk(const float* smem, int bufo,
                                              const float* zrow, int ch,
                                              int half, int m16, v8f acc[16]) {
  v2f af[4];
#pragma unroll
  for (int kk4 = 0; kk4 < 4; ++kk4)
    af[kk4] = *(const v2f*)(zrow + ch * 16 + kk4 * 4);

  const float* tb0 = smem + bufo + half * 32 + m16;
  v2f bf[16];
#pragma unroll
  for (int n = 0; n < 16; ++n) {
    bf[n].x = tb0[n * 256];
    bf[n].y = tb0[n * 256 + 16];
  }
#pragma unroll
  for (int kk4 = 0; kk4 < 4; ++kk4) {
#pragma unroll
    for (int n = 0; n < 16; ++n) {
      const v2f cur = bf[n];
      if (kk4 < 3) {
        bf[n].x = tb0[n * 256 + (kk4 + 1) * 64];
        bf[n].y = tb0[n * 256 + (kk4 + 1) * 64 + 16];
      }
      acc[n] = __builtin_amdgcn_wmma_f32_16x16x4_f32(
          false, af[kk4], false, cur, (short)0, acc[n], false, false);
    }
  }
}

// ---------------------------------------------------------------------------
// Kernel 2: q = z @ sigma_inv[c] (M=8192,K=256,N=256 per c) via fp32 WMMA,
// fused with d2[b,c] = sum_j (p[c,j] - q[b,c,j])^2.
// 128 threads (4 waves); wave w owns 16 b-rows; block owns one c.
// sigma_inv[c] double-buffered through LDS (TDM when available).
// ---------------------------------------------------------------------------
__global__ __launch_bounds__(128) void evo_gemm_d2(const float* __restrict__ z,
                                                   const float* __restrict__ sig,
                                                   const float* __restrict__ p,
                                                   float* __restrict__ d2) {
  constexpr int KC = 16;          // K rows per LDS chunk
  constexpr int BUF = KC * 256;   // 4096 dwords per buffer
  __shared__ float smem[2 * BUF];

  const int t = threadIdx.x;
  const int wave = t >> 5;
  const int lane = t & 31;
  const int half = lane >> 4;
  const int m16 = lane & 15;
  const int c = blockIdx.y;
  const int b0w = blockIdx.x * 64 + wave * 16;

  // A-matrix (16x4 fp32) lane layout: lanes 0-15 hold K=k,k+1; 16-31 K=k+2,k+3
  const float* zrow = z + (b0w + m16) * D_SZ + half * 2;
  const float* sigc = sig + c * (D_SZ * D_SZ);

  v8f acc[16];
  const v8f zero8 = {0.f, 0.f, 0.f, 0.f, 0.f, 0.f, 0.f, 0.f};
#pragma unroll
  for (int n = 0; n < 16; ++n) acc[n] = zero8;

#if USE_TDM
  const bool issuer = (t < 32);  // wave 0 drives the Tensor Data Mover
  if (issuer) {
    tdm_load_chunk(sigc, (unsigned)(uintptr_t)&smem[0]);
    __builtin_amdgcn_s_wait_tensorcnt((short)0);
  }
  __syncthreads();
  int bufo = 0;
#pragma unroll 1
  for (int ch = 0; ch < D_SZ / KC; ++ch) {
    if ((ch + 1 < D_SZ / KC) && issuer)
      tdm_load_chunk(sigc + (ch + 1) * KC * D_SZ,
                     (unsigned)(uintptr_t)&smem[bufo ^ BUF]);
    compute_chunk(smem, bufo, zrow, ch, half, m16, acc);
    if (issuer) __builtin_amdgcn_s_wait_tensorcnt((short)0);
    __syncthreads();
    bufo ^= BUF;
  }
#else
  // Fallback: register-staged double buffering. f indexes LDS float4s
  // sequentially (conflict-free stores); matching global float4 is
  // [k = (f>>2)&15][block n = f>>6][pos = f&3].
  float4 pre[8];
#pragma unroll
  for (int i = 0; i < 8; ++i) {
    const int f = i * 128 + t;
    pre[i] = ((const float4*)sigc)[((f >> 2) & 15) * 64 + (f >> 6) * 4 + (f & 3)];
  }
  int bufo = 0;
#pragma unroll 1
  for (int ch = 0; ch < D_SZ / KC; ++ch) {
#pragma unroll
    for (int i = 0; i < 8; ++i) {
      const int f = i * 128 + t;
      *(float4*)&smem[bufo + f * 4] = pre[i];
    }
    __syncthreads();
    if (ch + 1 < D_SZ / KC) {
      const float* nxt = sigc + (ch + 1) * KC * D_SZ;
#pragma unroll
      for (int i = 0; i < 8; ++i) {
        const int f = i * 128 + t;
        pre[i] = ((const float4*)nxt)[((f >> 2) & 15) * 64 + (f >> 6) * 4 + (f & 3)];
      }
    }
    compute_chunk(smem, bufo, zrow, ch, half, m16, acc);
    __syncthreads();
    bufo ^= BUF;
  }
#endif

  // epilogue: d2 = sum_j (p[c,j] - q[b,j])^2
  // acc[n] VGPR g @ lane l holds q[b0w + g + 8*half, n*16 + (l&15)]
  float s[8];
#pragma unroll
  for (int g = 0; g < 8; ++g) s[g] = 0.0f;
#pragma unroll
  for (int n = 0; n < 16; ++n) {
    const float pv = p[c * D_SZ + n * 16 + m16];
#pragma unroll
    for (int g = 0; g < 8; ++g) {
      const float diff = pv - acc[n][g];
      s[g] = fmaf(diff, diff, s[g]);
    }
  }
#pragma unroll
  for (int g = 0; g < 8; ++g) {
#pragma unroll
    for (int mask = 1; mask < 16; mask <<= 1)
      s[g] += __shfl_xor(s[g], mask, 32);
  }
  if (m16 == 0) {
#pragma unroll
    for (int g = 0; g < 8; ++g)
      d2[(b0w + g + 8 * half) * C_SZ + c] = s[g];
  }
}

// ---------------------------------------------------------------------------
// Kernel 3: softmax gate + ARX recurrence + psi-weighted combine.
// One lane per (b,c); 16-lane half-wave = one b.
// ---------------------------------------------------------------------------
__global__ __launch_bounds__(256) void evo_arx(const float* __restrict__ y,
                                               const float* __restrict__ u,
                                               const float* __restrict__ a_coef,
                                               const float* __restrict__ b_coef,
                                               const float* __restrict__ bias,
                                               const float* __restrict__ d2,
                                               float* __restrict__ out) {
  const int gt = blockIdx.x * 256 + threadIdx.x;
  const int b = gt >> 4;
  const int c = gt & 15;

  // softmax over c of -max(d2, 1e-8)
  const float x = -fmaxf(d2[b * C_SZ + c], 1e-8f);
  float mx = x;
#pragma unroll
  for (int mask = 1; mask < 16; mask <<= 1)
    mx = fmaxf(mx, __shfl_xor(mx, mask, 32));
  const float e = __expf(x - mx);
  float se = e;
#pragma unroll
  for (int mask = 1; mask < 16; mask <<= 1) se += __shfl_xor(se, mask, 32);
  const float psi = e / se;

  // ub = u[b,c,:] . b_coef[c,:] + bias[c]
  const float* ur = u + (b * C_SZ + c) * 32;
  const float* bc = b_coef + c * 32;
  float ub = bias[c];
#pragma unroll
  for (int k = 0; k < 32; ++k) ub = fmaf(ur[k], bc[k], ub);

  // ARX state: last 16 samples of y[b,c,:]
  float st[16], ac[16];
  const float* yr = y + (b * C_SZ + c) * 64 + 48;
#pragma unroll
  for (int o = 0; o < 16; ++o) {
    st[o] = yr[o];
    ac[o] = a_coef[c * 16 + o];
  }

  float* ob = out + b * 32;
#pragma unroll
  for (int l = 0; l < 32; ++l) {
    float yn = ub;
#pragma unroll
    for (int o = 0; o < 16; ++o) yn = fmaf(st[o], ac[o], yn);
#pragma unroll
    for (int o = 0; o < 15; ++o) st[o] = st[o + 1];
    st[15] = yn;
    float part = psi * yn;
#pragma unroll
    for (int mask = 1; mask < 16; mask <<= 1)
      part += __shfl_xor(part, mask, 32);
    if (c == 0) ob[l] = part;
  }
}

// ---------------------------------------------------------------------------
extern "C" void kernel_launch(void* const* d_in, const int* in_sizes, int n_in,
                              void* d_out, int out_size, void* d_ws,
                              size_t ws_size, hipStream_t stream) {
  const float* y = (const float*)d_in[0];        // (B,C,64)
  const float* z = (const float*)d_in[1];        // (B,1,256)
  const float* u = (const float*)d_in[2];        // (B,C,32)
  const float* mu = (const float*)d_in[3];       // (C,256)
  const float* sig = (const float*)d_in[4];      // (C,256,256)
  const float* a_coef = (const float*)d_in[5];   // (C,16)
  const float* b_coef = (const float*)d_in[6];   // (C,32)
  const float* bias = (const float*)d_in[7];     // (C,)
  float* out = (float*)d_out;                    // (B,1,32)

  float* wsf = (float*)d_ws;
  float* p = wsf;                 // C*D = 4096 floats
  float* d2 = wsf + C_SZ * D_SZ;  // B*C = 131072 floats

  evo_pvec<<<16, 256, 0, stream>>>(mu, sig, p);
  dim3 g2(B_SZ / 64, C_SZ);
  evo_gemm_d2<<<g2, 128, 0, stream>>>(z, sig, p, d2);
  evo_arx<<<(B_SZ * C_SZ) / 256, 256, 0, stream>>>(y, u, a_coef, b_coef, bias,
                                                   d2, out);
}